// GATConvCuGraph_66666482368959
// MI455X (gfx1250) — compile-verified
//
#include <hip/hip_runtime.h>
#include <hip/hip_bf16.h>

typedef float v2f __attribute__((ext_vector_type(2)));
typedef float v8f __attribute__((ext_vector_type(8)));

#define IN_CH    128
#define HEADS    4
#define OUT_CH   8
#define HC       32          // HEADS*OUT_CH
#define NEG_SLOPE 0.2f

// ---------------- init: out accumulator = 0, m = -inf (ordered int), s = 0 ----
__global__ void k_init(float* __restrict__ out, int* __restrict__ m_ord,
                       float* __restrict__ s, int N) {
  int tid = blockIdx.x * blockDim.x + threadIdx.x;
  if (tid < N * HC) out[tid] = 0.f;
  if (tid < N * HEADS) { m_ord[tid] = (int)0x80000000; s[tid] = 0.f; }
}

// ---------------- projection xh = x @ W^T via V_WMMA_F32_16X16X4_F32 ----------
// One wave -> one 16-row tile, both 16-col tiles. W staged in LDS.
__global__ void __launch_bounds__(256)
k_proj(const float* __restrict__ x, const float* __restrict__ W,
       float* __restrict__ xh, int N) {
  __shared__ float sW[HC * IN_CH];                 // 16 KB
  int tid = threadIdx.x;
  for (int i = tid; i < HC * IN_CH; i += 256) sW[i] = W[i];
  __syncthreads();

  int wave = tid >> 5;
  int lane = tid & 31;
  int tile = blockIdx.x * 8 + wave;                // index of 16-row tile
  int ntiles = N >> 4;                             // N % 16 handled by tail kernel
  if (tile >= ntiles) return;
  int m0  = tile << 4;
  int l15 = lane & 15;
  int hi  = lane >> 4;                             // lane-half selects K pair

  const float* xrow  = x  + (size_t)(m0 + l15) * IN_CH + hi * 2;
  const float* wrow0 = sW + (0  + l15) * IN_CH + hi * 2;
  const float* wrow1 = sW + (16 + l15) * IN_CH + hi * 2;

  v8f acc0 = {}; v8f acc1 = {};
  #pragma unroll
  for (int kc = 0; kc < IN_CH; kc += 4) {
    v2f a, b0, b1;
    a.x  = xrow[kc];   a.y  = xrow[kc + 1];        // A[M=l15][kc+2*hi .. +1]
    b0.x = wrow0[kc];  b0.y = wrow0[kc + 1];       // B[k][n] = W[n][k], n0=0
    b1.x = wrow1[kc];  b1.y = wrow1[kc + 1];       // n0=16
    acc0 = __builtin_amdgcn_wmma_f32_16x16x4_f32(false, a, false, b0,
                                                 (short)0, acc0, false, false);
    acc1 = __builtin_amdgcn_wmma_f32_16x16x4_f32(false, a, false, b1,
                                                 (short)0, acc1, false, false);
  }
  // D layout: VGPR r -> row r (lanes 0-15) / row r+8 (lanes 16-31), col = l15
  #pragma unroll
  for (int r = 0; r < 8; ++r) {
    int row = m0 + r + hi * 8;
    xh[(size_t)row * HC + l15]      = acc0[r];
    xh[(size_t)row * HC + 16 + l15] = acc1[r];
  }
}

// scalar tail for N % 16 rows (not hit for N=100000, kept for generality)
__global__ void k_proj_tail(const float* __restrict__ x, const float* __restrict__ W,
                            float* __restrict__ xh, int row0, int N) {
  int tid = blockIdx.x * blockDim.x + threadIdx.x;
  int nrem = N - row0;
  if (tid >= nrem * HC) return;
  int row = row0 + tid / HC, col = tid % HC;
  const float* xr = x + (size_t)row * IN_CH;
  const float* wr = W + (size_t)col * IN_CH;
  float acc = 0.f;
  for (int k = 0; k < IN_CH; ++k) acc += xr[k] * wr[k];
  xh[(size_t)row * HC + col] = acc;
}

// ---------------- per-node attention logits ----------------------------------
__global__ void k_att(const float* __restrict__ xh, const float* __restrict__ att,
                      float* __restrict__ a_src, float* __restrict__ a_dst, int N) {
  int tid = blockIdx.x * blockDim.x + threadIdx.x;
  if (tid >= N * HEADS) return;
  int n = tid >> 2, h = tid & 3;
  const float* v  = xh  + (size_t)n * HC + h * OUT_CH;
  const float* as = att + h * OUT_CH;
  const float* ad = att + HC + h * OUT_CH;
  float ssum = 0.f, dsum = 0.f;
  #pragma unroll
  for (int c = 0; c < OUT_CH; ++c) { ssum += v[c] * as[c]; dsum += v[c] * ad[c]; }
  a_src[tid] = ssum;
  a_dst[tid] = dsum;
}

// ordered-int mapping: monotone float -> int, enables atomicMax on floats
__device__ __forceinline__ int f2ord(float f) {
  int i = __float_as_int(f);
  return i >= 0 ? i : (i ^ 0x7FFFFFFF);
}
__device__ __forceinline__ float ord2f(int i) {
  return __int_as_float(i >= 0 ? i : (i ^ 0x7FFFFFFF));
}

// ---------------- edge pass 1: segmented max of leaky-relu logits -------------
__global__ void k_edge_max(const long long* __restrict__ ei,
                           const float* __restrict__ a_src,
                           const float* __restrict__ a_dst,
                           int* __restrict__ m_ord, int E) {
  int tid = blockIdx.x * blockDim.x + threadIdx.x;
  if (tid >= E * HEADS) return;
  int e = tid >> 2, h = tid & 3;
  int src = (int)ei[e];
  int dst = (int)ei[(size_t)E + e];
  float v = a_src[src * HEADS + h] + a_dst[dst * HEADS + h];
  v = (v >= 0.f) ? v : NEG_SLOPE * v;
  atomicMax(&m_ord[dst * HEADS + h], f2ord(v));
}

// ---------------- edge pass 2: fused exp-sum + unnormalized aggregation -------
__global__ void k_edge_acc(const long long* __restrict__ ei,
                           const float* __restrict__ a_src,
                           const float* __restrict__ a_dst,
                           const int* __restrict__ m_ord,
                           const float* __restrict__ xh,
                           float* __restrict__ s, float* __restrict__ out, int E) {
  int tid = blockIdx.x * blockDim.x + threadIdx.x;
  if (tid >= E * HEADS) return;
  int e = tid >> 2, h = tid & 3;
  int src = (int)ei[e];
  int dst = (int)ei[(size_t)E + e];
  float v = a_src[src * HEADS + h] + a_dst[dst * HEADS + h];
  v = (v >= 0.f) ? v : NEG_SLOPE * v;
  float m  = ord2f(m_ord[dst * HEADS + h]);
  float ex = __expf(v - m);
  atomicAdd(&s[dst * HEADS + h], ex);
  const float* xv = xh  + (size_t)src * HC + h * OUT_CH;   // 32B contiguous/lane
  float*       ov = out + (size_t)dst * HC + h * OUT_CH;
  #pragma unroll
  for (int c = 0; c < OUT_CH; ++c) atomicAdd(&ov[c], ex * xv[c]);
}

// ---------------- finalize: normalize + bias ---------------------------------
__global__ void k_final(float* __restrict__ out, const float* __restrict__ s,
                        const float* __restrict__ bias, int N) {
  int tid = blockIdx.x * blockDim.x + threadIdx.x;
  if (tid >= N * HC) return;
  int n = tid >> 5, j = tid & 31, h = j >> 3;
  float sv  = s[n * HEADS + h];
  float acc = out[tid];
  out[tid] = (sv > 0.f ? acc / sv : 0.f) + bias[j];
}

extern "C" void kernel_launch(void* const* d_in, const int* in_sizes, int n_in,
                              void* d_out, int out_size, void* d_ws, size_t ws_size,
                              hipStream_t stream) {
  const float*     x    = (const float*)d_in[0];
  const float*     W    = (const float*)d_in[1];
  const float*     att  = (const float*)d_in[2];
  const float*     bias = (const float*)d_in[3];
  const long long* ei   = (const long long*)d_in[4];
  int N = in_sizes[0] / IN_CH;
  int E = in_sizes[4] / 2;
  float* out = (float*)d_out;

  // workspace carve-up (~19.2 MB total)
  float* xh    = (float*)d_ws;                       // N*32
  float* a_src = xh + (size_t)N * HC;                // N*4
  float* a_dst = a_src + (size_t)N * HEADS;          // N*4
  int*   m_ord = (int*)(a_dst + (size_t)N * HEADS);  // N*4
  float* s     = (float*)(m_ord + (size_t)N * HEADS);// N*4

  const int B = 256;
  k_init<<<(N * HC + B - 1) / B, B, 0, stream>>>(out, m_ord, s, N);

  int ntiles = N >> 4;
  if (ntiles > 0)
    k_proj<<<(ntiles + 7) / 8, B, 0, stream>>>(x, W, xh, N);
  int rem = N - (ntiles << 4);
  if (rem > 0)
    k_proj_tail<<<(rem * HC + B - 1) / B, B, 0, stream>>>(x, W, xh, ntiles << 4, N);

  k_att<<<(N * HEADS + B - 1) / B, B, 0, stream>>>(xh, att, a_src, a_dst, N);
  k_edge_max<<<(E * HEADS + B - 1) / B, B, 0, stream>>>(ei, a_src, a_dst, m_ord, E);
  k_edge_acc<<<(E * HEADS + B - 1) / B, B, 0, stream>>>(ei, a_src, a_dst, m_ord,
                                                        xh, s, out, E);
  k_final<<<(N * HC + B - 1) / B, B, 0, stream>>>(out, s, bias, N);
}